// TimeSeriesODENet_26182120636916
// MI455X (gfx1250) — compile-verified
//
#include <hip/hip_runtime.h>
#include <hip/hip_bf16.h>

#define TT 512
#define BB 256
#define CC 65
#define HID 128

typedef __attribute__((ext_vector_type(2))) float v2f;
typedef __attribute__((ext_vector_type(8))) float v8f;

// ---------------- LDS layout (float offsets, all even) ----------------
// Weights stored per-wave-n-tile, k-pair interleaved:
//   flat = ((w*KK + kk)*16 + nn)*2 + p   where k = 2*kk+p, n = 16*w+nn, KK = Krows/2
// h / htmp / x stored k-pair interleaved: flat = kk*32 + 2*m + p  (k = 2*kk+p)
#define OFF_WR    0        // 160 rows  -> 8*80*16*2 = 20480 floats
#define OFF_WZ    20480    // 160 rows  -> 20480
#define OFF_WHN   40960    // 128 rows  -> 16384
#define OFF_WODE  57344    // 128 rows  -> 16384
#define OFF_H     73728    // 16x128    -> 2048
#define OFF_HT    75776    // 16x128    -> 2048
#define OFF_X     77824    // 16x32     -> 512
#define OFF_DT    78336    // 16
#define OFF_OBS   78352    // 16
#define OFF_TIMES 78368    // 512
#define OFF_ANY   78880    // 512
#define LDS_FLOATS 79392   // 317,568 bytes < 320 KB

__device__ inline v8f zero8() { v8f z = {0.f,0.f,0.f,0.f,0.f,0.f,0.f,0.f}; return z; }

__device__ inline v8f wmma4(v2f a, v2f b, v8f c) {
  // V_WMMA_F32_16X16X4_F32 : D = A(16x4) x B(4x16) + C(16x16), all f32
  return __builtin_amdgcn_wmma_f32_16x16x4_f32(false, a, false, b, (short)0, c, false, false);
}

// matmul over WMMA K-blocks [KB0, KB1); A read from interleaved buffer at abase
// using A-local block index (kb-KB0); B read from weight tile at wtile using kb.
template<int KB0, int KB1>
__device__ inline v8f mm_range(const float* s, int abase, int wtile, int half, int mn, v8f acc) {
#pragma unroll
  for (int kb = KB0; kb < KB1; ++kb) {
    v2f a = *(const v2f*)(s + abase + (((kb - KB0) * 2 + half) * 32 + 2 * mn));
    v2f b = *(const v2f*)(s + wtile + ((kb * 2 + half) * 16 + mn) * 2);
    acc = wmma4(a, b, acc);
  }
  return acc;
}

// Write this wave's 16-column D-tile slice into an interleaved [kk][m][p] buffer.
__device__ inline void store_cols(float* s, int base, int ncol, int half, v8f v) {
  int off = base + (ncol >> 1) * 32 + (ncol & 1);
#pragma unroll
  for (int i = 0; i < 8; ++i) s[off + 2 * (i + 8 * half)] = v[i];
}

__device__ inline float fast_tanh(float x) {
  x = fminf(fmaxf(x, -15.f), 15.f);
  float e = __expf(2.f * x);
  return (e - 1.f) / (e + 1.f);
}
__device__ inline float fast_sig(float x) { return 1.f / (1.f + __expf(-x)); }

__device__ inline void load_w(float* dst, const float* __restrict__ Wg, int Krows, int tid) {
  const int KK = Krows >> 1;
  const int total = Krows * HID;
  for (int g = tid; g < total; g += 256) {
    int k = g >> 7, n = g & 127;
    int wv = n >> 4, nn = n & 15;
    int kk = k >> 1, p = k & 1;
    dst[((wv * KK + kk) * 16 + nn) * 2 + p] = Wg[g];
  }
}

// ---------------- precompute: spline eval + transforms + obs masks ----------------
__device__ inline float raw_spline(const float* __restrict__ times, const float* __restrict__ a,
                                   const float* __restrict__ b, const float* __restrict__ c2,
                                   const float* __restrict__ d, int bat, int k, int ch) {
  int idx = (k > 0) ? (k - 1) : 0;           // clip(k-1, 0, T-2)
  float fr = times[k] - times[idx];
  size_t o = ((size_t)bat * (TT - 1) + idx) * CC + ch;
  float inner = 0.5f * c2[o] + d[o] * (fr * (1.f / 3.f));
  inner = b[o] + inner * fr;
  return a[o] + inner * fr;
}

__global__ void init_any_kernel(unsigned* __restrict__ anyg) {
  int i = threadIdx.x;
  if (i < TT) anyg[i] = 0u;
}

__global__ void precompute_kernel(const float* __restrict__ times, const float* __restrict__ ca,
                                  const float* __restrict__ cb, const float* __restrict__ cc2,
                                  const float* __restrict__ cd, float* __restrict__ feat,
                                  float* __restrict__ x0g, float* __restrict__ obsg,
                                  unsigned* __restrict__ anyg) {
  const int sIdx = blockIdx.x;   // timestep
  const int b = blockIdx.y;      // batch row
  const int c = threadIdx.x;     // channel (0..95; active < 65)
  __shared__ float red[32];
  if (c < CC) {
    float v = raw_spline(times, ca, cb, cc2, cd, b, sIdx, c);
    if (c == 0) {
      float sub = (sIdx == 0) ? times[0] : times[sIdx - 1];
      x0g[(size_t)sIdx * BB + b] = v - sub;
    } else if (c <= 32) {
      float prev = (sIdx == 0) ? 0.f : raw_spline(times, ca, cb, cc2, cd, b, sIdx - 1, c);
      red[c - 1] = v - prev;   // diffed indicator channel
    } else {
      feat[((size_t)sIdx * BB + b) * 32 + (c - 33)] = v;   // GRU feature channels
    }
  }
  __syncthreads();
  if (c == 0) {
    float mx = red[0];
#pragma unroll
    for (int i = 1; i < 32; ++i) mx = fmaxf(mx, red[i]);
    float ob = (mx > 0.5f) ? 1.f : 0.f;
    obsg[(size_t)sIdx * BB + b] = ob;
    if (ob > 0.5f) atomicOr(&anyg[sIdx], 1u);
  }
}

// ---------------- persistent GRU-ODE scan: 16 WGs x (16 rows, 8 waves) ----------------
__global__ __launch_bounds__(256, 1)
void scan_kernel(const float* __restrict__ times, const float* __restrict__ feat,
                 const float* __restrict__ x0g, const float* __restrict__ obsg,
                 const unsigned* __restrict__ anyg, const int* __restrict__ fidx,
                 const float* __restrict__ Wr, const float* __restrict__ br,
                 const float* __restrict__ Wz, const float* __restrict__ bz,
                 const float* __restrict__ Wxn, const float* __restrict__ bxn,
                 const float* __restrict__ Whn, const float* __restrict__ bhn,
                 const float* __restrict__ Wode, const float* __restrict__ bode,
                 float* __restrict__ hfin) {
  __shared__ float s[LDS_FLOATS];
  const int tid = threadIdx.x;
  const int lane = tid & 31;
  const int w = tid >> 5;          // wave id 0..7 -> owns hidden cols 16w..16w+15
  const int half = lane >> 4;      // 0: lanes 0-15, 1: lanes 16-31
  const int mn = lane & 15;        // A-row index / B-D column index
  const int b0 = blockIdx.x * 16;  // batch tile
  const int ncol = w * 16 + mn;

  // ---- preamble ----
  load_w(s + OFF_WR, Wr, 160, tid);
  load_w(s + OFF_WZ, Wz, 160, tid);
  load_w(s + OFF_WHN, Whn, 128, tid);
  load_w(s + OFF_WODE, Wode, 128, tid);
  for (int i = tid; i < 2048; i += 256) s[OFF_H + i] = 0.f;
  for (int i = tid; i < TT; i += 256) {
    s[OFF_TIMES + i] = times[i];
    s[OFF_ANY + i] = anyg[i] ? 1.f : 0.f;
  }
  if (tid < 16) s[OFF_DT + tid] = 0.f;

  // Wxn kept in registers (B-operand layout), saves 16 KB of LDS
  v2f wxnr[8];
#pragma unroll
  for (int kb = 0; kb < 8; ++kb) {
    int k = kb * 4 + half * 2;
    v2f t2; t2[0] = Wxn[k * HID + ncol]; t2[1] = Wxn[(k + 1) * HID + ncol];
    wxnr[kb] = t2;
  }
  const float brv = br[ncol], bzv = bz[ncol], bxnv = bxn[ncol], bhnv = bhn[ncol], bov = bode[ncol];
  int fir[8];
#pragma unroll
  for (int i = 0; i < 8; ++i) fir[i] = fidx[b0 + i + 8 * half];

  const int wr_t = OFF_WR + w * 2560;
  const int wz_t = OFF_WZ + w * 2560;
  const int whn_t = OFF_WHN + w * 2048;
  const int wode_t = OFF_WODE + w * 2048;

  v8f h = zero8();   // this wave's slice of the hidden state: rows 0..15 x cols 16w..16w+15
  __syncthreads();

  for (int t = 0; t < TT; ++t) {
    if (t > 0) {
      const float td = s[OFF_TIMES + t] - s[OFF_TIMES + t - 1];
      v8f k1, k2, k3, k4, tmp;
      v8f d = mm_range<0, 32>(s, OFF_H, wode_t, half, mn, zero8());
#pragma unroll
      for (int i = 0; i < 8; ++i) k1[i] = fast_tanh(d[i] + bov);
#pragma unroll
      for (int i = 0; i < 8; ++i) tmp[i] = h[i] + 0.5f * td * k1[i];
      store_cols(s, OFF_HT, ncol, half, tmp);
      __syncthreads();
      d = mm_range<0, 32>(s, OFF_HT, wode_t, half, mn, zero8());
#pragma unroll
      for (int i = 0; i < 8; ++i) k2[i] = fast_tanh(d[i] + bov);
      __syncthreads();
#pragma unroll
      for (int i = 0; i < 8; ++i) tmp[i] = h[i] + 0.5f * td * k2[i];
      store_cols(s, OFF_HT, ncol, half, tmp);
      __syncthreads();
      d = mm_range<0, 32>(s, OFF_HT, wode_t, half, mn, zero8());
#pragma unroll
      for (int i = 0; i < 8; ++i) k3[i] = fast_tanh(d[i] + bov);
      __syncthreads();
#pragma unroll
      for (int i = 0; i < 8; ++i) tmp[i] = h[i] + td * k3[i];
      store_cols(s, OFF_HT, ncol, half, tmp);
      __syncthreads();
      d = mm_range<0, 32>(s, OFF_HT, wode_t, half, mn, zero8());
#pragma unroll
      for (int i = 0; i < 8; ++i) k4[i] = fast_tanh(d[i] + bov);
#pragma unroll
      for (int i = 0; i < 8; ++i)
        h[i] = h[i] + (td * (1.f / 6.f)) * (k1[i] + 2.f * k2[i] + 2.f * k3[i] + k4[i]);
      store_cols(s, OFF_H, ncol, half, h);   // OFF_H only read again after next sync
    }

    const bool anyo = s[OFF_ANY + t] > 0.5f;   // workgroup-uniform
    if (anyo) {
      // stage GRU inputs: x (with +dt on feature 0), obs flags; update dt
      for (int idx = tid; idx < 512; idx += 256) {
        int m = idx >> 5, k = idx & 31;
        size_t row = (size_t)t * BB + b0 + m;
        float v = feat[row * 32 + k];
        if (k == 0) {
          v += s[OFF_DT + m];
          float ob = obsg[row];
          s[OFF_OBS + m] = ob;
          if (ob < 0.5f) s[OFF_DT + m] += x0g[row];   // dt += Xi[:,0] where !obs
        }
        s[OFF_X + (k >> 1) * 32 + 2 * m + (k & 1)] = v;
      }
    }
    __syncthreads();
    if (anyo) {
      // r,z over comb=[x(32), h(128)]  (K blocks 0..7 from x, 8..39 from h)
      v8f rpre = mm_range<0, 8>(s, OFF_X, wr_t, half, mn, zero8());
      rpre = mm_range<8, 40>(s, OFF_H, wr_t, half, mn, rpre);
      v8f zpre = mm_range<0, 8>(s, OFF_X, wz_t, half, mn, zero8());
      zpre = mm_range<8, 40>(s, OFF_H, wz_t, half, mn, zpre);
      v8f gpre = mm_range<0, 32>(s, OFF_H, whn_t, half, mn, zero8());
      v8f xpre = zero8();
#pragma unroll
      for (int kb = 0; kb < 8; ++kb) {
        v2f a = *(const v2f*)(s + OFF_X + ((kb * 2 + half) * 32 + 2 * mn));
        xpre = wmma4(a, wxnr[kb], xpre);
      }
#pragma unroll
      for (int i = 0; i < 8; ++i) {
        float r = fast_sig(rpre[i] + brv);
        float z = fast_sig(zpre[i] + bzv);
        float n = fast_tanh(xpre[i] + bxnv + r * (gpre[i] + bhnv));
        float nh = (1.f - z) * n + z * h[i];
        float ob = s[OFF_OBS + i + 8 * half];
        h[i] = (ob > 0.5f) ? nh : h[i];
      }
      __syncthreads();                        // all waves done reading OFF_H
      store_cols(s, OFF_H, ncol, half, h);
    }
    // capture rows whose final_index == t
#pragma unroll
    for (int i = 0; i < 8; ++i)
      if (fir[i] == t) hfin[(size_t)(b0 + i + 8 * half) * HID + ncol] = h[i];
    __syncthreads();   // OFF_H/OFF_HT visible & quiesced before next step
  }
}

// ---------------- final linear layer ----------------
__global__ void out_kernel(const float* __restrict__ hfin, const float* __restrict__ Wl,
                           const float* __restrict__ bl, float* __restrict__ out) {
  int i = blockIdx.x * blockDim.x + threadIdx.x;
  if (i >= BB * 10) return;
  int b = i / 10, o = i % 10;
  float acc = bl[o];
#pragma unroll 8
  for (int k = 0; k < HID; ++k) acc += hfin[b * HID + k] * Wl[k * 10 + o];
  out[i] = acc;
}

extern "C" void kernel_launch(void* const* d_in, const int* in_sizes, int n_in,
                              void* d_out, int out_size, void* d_ws, size_t ws_size,
                              hipStream_t stream) {
  const float* times = (const float*)d_in[0];
  const float* ca    = (const float*)d_in[1];
  const float* cb    = (const float*)d_in[2];
  const float* cc2   = (const float*)d_in[3];
  const float* cd    = (const float*)d_in[4];
  const int*   fidx  = (const int*)d_in[5];
  const float* Wr    = (const float*)d_in[6];
  const float* br    = (const float*)d_in[7];
  const float* Wz    = (const float*)d_in[8];
  const float* bz    = (const float*)d_in[9];
  const float* Wxn   = (const float*)d_in[10];
  const float* bxn   = (const float*)d_in[11];
  const float* Whn   = (const float*)d_in[12];
  const float* bhn   = (const float*)d_in[13];
  const float* Wode  = (const float*)d_in[14];
  const float* bode  = (const float*)d_in[15];
  const float* Wlin  = (const float*)d_in[16];
  const float* blin  = (const float*)d_in[17];
  float* out = (float*)d_out;

  char* ws = (char*)d_ws;
  const size_t FEAT_B = (size_t)TT * BB * 32 * 4;   // 16 MiB
  const size_t COL_B  = (size_t)TT * BB * 4;        // 512 KiB
  float*    feat = (float*)ws;
  float*    x0g  = (float*)(ws + FEAT_B);
  float*    obsg = (float*)(ws + FEAT_B + COL_B);
  unsigned* anyg = (unsigned*)(ws + FEAT_B + 2 * COL_B);
  float*    hfin = (float*)(ws + FEAT_B + 2 * COL_B + 2048);

  init_any_kernel<<<1, 512, 0, stream>>>(anyg);
  precompute_kernel<<<dim3(TT, BB), 96, 0, stream>>>(times, ca, cb, cc2, cd,
                                                     feat, x0g, obsg, anyg);
  scan_kernel<<<BB / 16, 256, 0, stream>>>(times, feat, x0g, obsg, anyg, fidx,
                                           Wr, br, Wz, bz, Wxn, bxn, Whn, bhn,
                                           Wode, bode, hfin);
  out_kernel<<<(BB * 10 + 255) / 256, 256, 0, stream>>>(hfin, Wlin, blin, out);
}